// Encoder_44324062494985
// MI455X (gfx1250) — compile-verified
//
#include <hip/hip_runtime.h>

typedef __attribute__((ext_vector_type(2))) float v2f;
typedef __attribute__((ext_vector_type(8))) float v8f;

#define IN_F   512
#define OUT_F  256
#define ALPHA_ 0.15f
#define ONEMA_ 0.85f
#define SCALE_ 1.8f
#define EPS_   1e-12f

// ---------- degree / norm precompute ----------
__global__ void deg_init_kernel(float* __restrict__ deg, int N) {
    int i = blockIdx.x * blockDim.x + threadIdx.x;
    if (i < N) deg[i] = 1.0f;                 // self-loop contributes 1
}

__global__ void deg_acc_kernel(const int* __restrict__ dst, float* __restrict__ deg, int E) {
    int i = blockIdx.x * blockDim.x + threadIdx.x;
    if (i < E) unsafeAtomicAdd(&deg[dst[i]], 1.0f);
}

__global__ void dinv_kernel(float* __restrict__ deg, int N) {
    int i = blockIdx.x * blockDim.x + threadIdx.x;
    if (i < N) deg[i] = rsqrtf(deg[i]);       // deg >= 1 always (self loops)
}

__global__ void snorm_kernel(const int* __restrict__ src, const int* __restrict__ dst,
                             const float* __restrict__ dinv, float* __restrict__ snorm, int E) {
    int i = blockIdx.x * blockDim.x + threadIdx.x;
    if (i < E) snorm[i] = ONEMA_ * dinv[src[i]] * dinv[dst[i]];
}

// ---------- fused linear + L2-normalize (WMMA f32 16x16x4) ----------
// block = 128 threads (4 waves). Block computes rows [16*bid, 16*bid+16) x all 256 cols.
// Wave w computes a 16x64 slab (4 WMMA n-tiles). Row-wise sum-of-squares reduced in LDS.
__global__ __launch_bounds__(128)
void linear_norm_kernel(const float* __restrict__ x, const float* __restrict__ W,
                        const float* __restrict__ b, float* __restrict__ h, int N) {
    __shared__ float s_ssq[16];

    const int lane  = threadIdx.x & 31;
    const int wave  = threadIdx.x >> 5;
    const int lo    = lane & 15;    // row within tile (A: M, B: N)
    const int hi    = lane >> 4;    // selects K pair {0,1} vs {2,3}
    const int mBase = blockIdx.x << 4;
    const int nBase = wave << 6;

    v8f zero = {};
    v8f acc[4] = {zero, zero, zero, zero};

    // A frag: a[j] = x[mBase+lo][k + 2*hi + j]
    // B frag: b[j] = W1T[k+2*hi+j][nBase+t*16+lo] = W1[nBase+t*16+lo][k+2*hi+j]
    const float* ap  = x + (size_t)(mBase + lo) * IN_F + (hi << 1);
    const float* bp  = W + (size_t)(nBase + lo) * IN_F + (hi << 1);

    for (int k = 0; k < IN_F; k += 4) {
        v2f a = *(const v2f*)(ap + k);
#pragma unroll
        for (int t = 0; t < 4; ++t) {
            v2f bv = *(const v2f*)(bp + (size_t)(t * 16) * IN_F + k);
            acc[t] = __builtin_amdgcn_wmma_f32_16x16x4_f32(
                false, a, false, bv, (short)0, acc[t], false, false);
        }
    }

    // add bias; accumulate per-row (m = hi*8 + r) partial sum of squares
    float part[8];
#pragma unroll
    for (int r = 0; r < 8; ++r) part[r] = 0.0f;
#pragma unroll
    for (int t = 0; t < 4; ++t) {
        float bias = b[nBase + (t << 4) + lo];
#pragma unroll
        for (int r = 0; r < 8; ++r) {
            float v = acc[t][r] + bias;
            acc[t][r] = v;
            part[r] += v * v;
        }
    }

    if (threadIdx.x < 16) s_ssq[threadIdx.x] = 0.0f;
    __syncthreads();
#pragma unroll
    for (int r = 0; r < 8; ++r) atomicAdd(&s_ssq[(hi << 3) + r], part[r]);
    __syncthreads();

    float sc[8];
#pragma unroll
    for (int r = 0; r < 8; ++r) {
        float nrm = sqrtf(s_ssq[(hi << 3) + r]);
        sc[r] = SCALE_ / fmaxf(nrm, EPS_);
    }
#pragma unroll
    for (int t = 0; t < 4; ++t)
#pragma unroll
        for (int r = 0; r < 8; ++r)
            h[(size_t)(mBase + (hi << 3) + r) * OUT_F + nBase + (t << 4) + lo] =
                acc[t][r] * sc[r];
}

// ---------- APPNP step: init  zn = alpha*h0 + 0.85*dinv^2*z  (vectorized float4) ----------
__global__ void prop_init_kernel(const float* __restrict__ h0, const float* __restrict__ z,
                                 const float* __restrict__ dinv, float* __restrict__ zn,
                                 int total4) {
    int i = blockIdx.x * blockDim.x + threadIdx.x;
    if (i >= total4) return;
    int n = i >> 6;                           // 64 float4 per node row
    float di = dinv[n];
    float sf = ONEMA_ * di * di;
    float4 hv = ((const float4*)h0)[i];
    float4 zv = ((const float4*)z)[i];
    float4 o;
    o.x = ALPHA_ * hv.x + sf * zv.x;
    o.y = ALPHA_ * hv.y + sf * zv.y;
    o.z = ALPHA_ * hv.z + sf * zv.z;
    o.w = ALPHA_ * hv.w + sf * zv.w;
    ((float4*)zn)[i] = o;
}

// ---------- APPNP step: edge scatter  zn[dst] += snorm[e]*z[src]  (1 wave / edge) ----------
__global__ __launch_bounds__(256)
void edge_kernel(const float* __restrict__ z, const int* __restrict__ src,
                 const int* __restrict__ dst, const float* __restrict__ w,
                 float* __restrict__ zn, int E) {
    int e    = blockIdx.x * 8 + (threadIdx.x >> 5);
    int lane = threadIdx.x & 31;
    if (e >= E) return;
    int   s = src[e];
    int   d = dst[e];
    float c = w[e];

    const float4* zp = ((const float4*)(z + (size_t)s * OUT_F)) + (lane << 1);
    float4 v0 = zp[0];
    float4 v1 = zp[1];
    float* op = zn + (size_t)d * OUT_F + (lane << 3);
    unsafeAtomicAdd(op + 0, c * v0.x);
    unsafeAtomicAdd(op + 1, c * v0.y);
    unsafeAtomicAdd(op + 2, c * v0.z);
    unsafeAtomicAdd(op + 3, c * v0.w);
    unsafeAtomicAdd(op + 4, c * v1.x);
    unsafeAtomicAdd(op + 5, c * v1.y);
    unsafeAtomicAdd(op + 6, c * v1.z);
    unsafeAtomicAdd(op + 7, c * v1.w);
}

extern "C" void kernel_launch(void* const* d_in, const int* in_sizes, int n_in,
                              void* d_out, int out_size, void* d_ws, size_t ws_size,
                              hipStream_t stream) {
    const float* x   = (const float*)d_in[0];
    const int*   ei  = (const int*)d_in[1];   // [2, E] int
    const float* W1  = (const float*)d_in[2]; // [256, 512]
    const float* b1  = (const float*)d_in[3]; // [256]
    float*       out = (float*)d_out;

    const int N = in_sizes[0] / IN_F;
    const int E = in_sizes[1] / 2;
    const int* srcp = ei;
    const int* dstp = ei + E;

    // workspace layout (256B-aligned sections)
    float* wsf   = (float*)d_ws;
    size_t Npad  = ((size_t)N + 63) & ~(size_t)63;
    size_t Epad  = ((size_t)E + 63) & ~(size_t)63;
    float* dinv  = wsf;                         // N
    float* snorm = wsf + Npad;                  // E
    float* h0    = snorm + Epad;                // N*256
    float* ping  = h0 + (size_t)N * OUT_F;      // N*256

    // degrees -> dinv -> per-edge weights
    deg_init_kernel<<<(N + 255) / 256, 256, 0, stream>>>(dinv, N);
    deg_acc_kernel<<<(E + 255) / 256, 256, 0, stream>>>(dstp, dinv, E);
    dinv_kernel<<<(N + 255) / 256, 256, 0, stream>>>(dinv, N);
    snorm_kernel<<<(E + 255) / 256, 256, 0, stream>>>(srcp, dstp, dinv, snorm, E);

    // h0 = scale * l2norm(x @ W1^T + b1)
    linear_norm_kernel<<<N / 16, 128, 0, stream>>>(x, W1, b1, h0, N);

    // APPNP K=10; odd steps write ping, even steps write d_out (step 10 -> d_out)
    const int total4     = N * (OUT_F / 4);
    const int initBlocks = (total4 + 255) / 256;
    const int edgeBlocks = (E + 7) / 8;
    for (int k = 1; k <= 10; ++k) {
        const float* zin  = (k == 1) ? h0 : ((k & 1) ? out : ping);
        float*       zout = (k & 1) ? ping : out;
        prop_init_kernel<<<initBlocks, 256, 0, stream>>>(h0, zin, dinv, zout, total4);
        edge_kernel<<<edgeBlocks, 256, 0, stream>>>(zin, srcp, dstp, snorm, zout, E);
    }
}